// Decoder_68315749810368
// MI455X (gfx1250) — compile-verified
//
#include <hip/hip_runtime.h>
#include <math.h>

typedef __attribute__((ext_vector_type(2))) float v2f;
typedef __attribute__((ext_vector_type(8))) float v8f;

#define BN_INV 0.99999500003749972f   /* 1/sqrt(1+1e-5) */

__device__ __forceinline__ float sigmoidf_(float x){ return 1.0f/(1.0f+__expf(-x)); }
__device__ __forceinline__ float softplusf_(float x){ return (x>20.f)? x : log1pf(__expf(x)); }

// ======================================================================
// Generic WMMA GEMM: out[m,n] = act( sum_k A[m,k]*W[n,k] + bias[n] )
// A: (M x lda) row-major (uses first K cols), W: (N x K) row-major,
// out: (M x ldo). One wave per 16x16 tile, V_WMMA_F32_16X16X4_F32.
// Fragment mapping (wave32):
//   A: a[v] = A[m0 + (lane&15)][k0 + 2*(lane>>4) + v]
//   B: b[v] = W[n0 + (lane&15)][k0 + 2*(lane>>4) + v]   (B[k][n] = W[n][k])
//   D: d[i] = out[m0 + i + 8*(lane>>4)][n0 + (lane&15)]
// ======================================================================
__global__ __launch_bounds__(128) void gemm_xwt(
    const float* __restrict__ A, const float* __restrict__ W,
    const float* __restrict__ bias, float* __restrict__ out,
    int M, int N, int K, int lda, int ldo, int act)
{
  const int lane = threadIdx.x & 31;
  const int wave = threadIdx.x >> 5;
  const int r    = lane & 15;
  const int g    = lane >> 4;
  const int m0   = (blockIdx.y * 4 + wave) << 4;
  const int n0   = blockIdx.x << 4;
  const bool nok = (n0 + r) < N;
  const float* Arow = A + (size_t)(m0 + r) * lda;
  const float* Wrow = W + (size_t)(nok ? (n0 + r) : 0) * K;  // safe base for edge lanes
  v8f acc = {0.f,0.f,0.f,0.f,0.f,0.f,0.f,0.f};
  for (int k = 0; k < K; k += 4) {
    const int ka = k + 2*g;
    v2f av, bv;
    av.x = Arow[ka];
    av.y = Arow[ka+1];
    float b0 = Wrow[ka];
    float b1 = Wrow[ka+1];
    bv.x = nok ? b0 : 0.f;
    bv.y = nok ? b1 : 0.f;
    acc = __builtin_amdgcn_wmma_f32_16x16x4_f32(
        /*neg_a=*/false, av, /*neg_b=*/false, bv,
        /*c_mod=*/(short)0, acc, /*reuse_a=*/false, /*reuse_b=*/false);
  }
  if (!nok) return;                       // after all WMMAs (EXEC full during WMMA)
  const float bb = bias ? bias[n0 + r] : 0.f;
#pragma unroll
  for (int i = 0; i < 8; ++i) {
    float v = acc[i] + bb;
    if (act == 1) v = softplusf_(v);
    out[(size_t)(m0 + i + 8*g) * ldo + (n0 + r)] = v;
  }
}

// ---------- NCHW -> (B, HW, C) sequence layout -------------------------
__global__ void nchw_to_seq(const float* __restrict__ in, float* __restrict__ out,
                            int C, int HW, int total)
{
  int idx = blockIdx.x * blockDim.x + threadIdx.x;
  if (idx >= total) return;
  int c = idx % C;
  int p = (idx / C) % HW;
  int b = idx / (C * HW);
  out[idx] = in[((size_t)b * C + c) * HW + p];
}

// ---------- pack up-conv weight (C,O,2,2) -> Wp[(ij*O+o), c] -----------
__global__ void pack_upw(const float* __restrict__ w, float* __restrict__ wp,
                         int C, int O, int total)
{
  int idx = blockIdx.x * blockDim.x + threadIdx.x;
  if (idx >= total) return;
  int c  = idx % C;
  int o  = (idx / C) % O;
  int ij = idx / (C * O);
  wp[idx] = w[((size_t)c * O + o) * 4 + ij];
}

// ---------- scatter up-conv GEMM result + bias + NCHW skip -> seq ------
__global__ void up_scatter(const float* __restrict__ tmp, const float* __restrict__ skip,
                           const float* __restrict__ bias, float* __restrict__ out,
                           int Hi, int Wi, int O, int total)
{
  int idx = blockIdx.x * blockDim.x + threadIdx.x;
  if (idx >= total) return;
  const int Ho = 2*Hi, Wo = 2*Wi;
  int o  = idx % O;
  int p  = (idx / O) % (Ho * Wo);
  int b  = idx / (O * Ho * Wo);
  int ho = p / Wo, wo = p % Wo;
  int h = ho >> 1, i = ho & 1, w = wo >> 1, j = wo & 1;
  float v = tmp[((size_t)b * Hi * Wi + h * Wi + w) * (4*O) + (i*2 + j) * O + o];
  v += bias[o];
  v += skip[(((size_t)b * O + o) * Ho + ho) * Wo + wo];
  out[idx] = v;
}

// ---------- per-batch mean / rstd over all C*H*W -----------------------
__global__ void norm_stats(const float* __restrict__ x, float* __restrict__ stats, int LC)
{
  __shared__ float ssum[256], ssq[256];
  const int b = blockIdx.x;
  const float* p = x + (size_t)b * LC;
  float s = 0.f, q = 0.f;
  for (int i = threadIdx.x; i < LC; i += blockDim.x) { float v = p[i]; s += v; q += v*v; }
  ssum[threadIdx.x] = s; ssq[threadIdx.x] = q;
  __syncthreads();
  for (int off = 128; off > 0; off >>= 1) {
    if (threadIdx.x < off) { ssum[threadIdx.x] += ssum[threadIdx.x + off];
                             ssq[threadIdx.x]  += ssq[threadIdx.x + off]; }
    __syncthreads();
  }
  if (threadIdx.x == 0) {
    float mu  = ssum[0] / (float)LC;
    float var = ssq[0] / (float)LC - mu * mu;
    stats[b*2]   = mu;
    stats[b*2+1] = rsqrtf(var + 1e-5f);
  }
}

// ---------- xn = clip( (x-mu)*rstd*g[c] + b[c], -10, 10 ) --------------
__global__ void gn_clip(const float* __restrict__ x, const float* __restrict__ stats,
                        const float* __restrict__ g, const float* __restrict__ bb,
                        float* __restrict__ out, int C, int LC, int total)
{
  int idx = blockIdx.x * blockDim.x + threadIdx.x;
  if (idx >= total) return;
  int c = idx % C;
  int b = idx / LC;
  float v = (x[idx] - stats[b*2]) * stats[b*2+1] * g[c] + bb[c];
  v = fminf(fmaxf(v, -10.f), 10.f);
  out[idx] = v;
}

// ---------- depthwise causal conv1d (k=4) + bias + SiLU ----------------
__global__ void conv1d_silu(const float* __restrict__ xz, const float* __restrict__ w,
                            const float* __restrict__ cb, float* __restrict__ out,
                            int L, int D, int ldin, int total)
{
  int idx = blockIdx.x * blockDim.x + threadIdx.x;
  if (idx >= total) return;
  int d = idx % D;
  int l = (idx / D) % L;
  int b = idx / (D * L);
  float acc = cb[d];
#pragma unroll
  for (int t = 0; t < 4; ++t) {
    int ls = l - 3 + t;
    if (ls >= 0) acc += xz[((size_t)b * L + ls) * ldin + d] * w[d*4 + t];
  }
  out[idx] = acc * sigmoidf_(acc);     // silu
}

// ---------- selective scan, fused (+x*D)*silu(z) -----------------------
// one thread per (b,d); 16 states in registers; B/C vectors via LDS
__global__ void mamba_scan(const float* __restrict__ dt, const float* __restrict__ dbl,
                           const float* __restrict__ xc, const float* __restrict__ z,
                           const float* __restrict__ A_log, const float* __restrict__ Dv,
                           float* __restrict__ yout,
                           int L, int D, int dtrank, int lddbl, int ldz)
{
  __shared__ float sB[16], sC[16];
  const int b = blockIdx.x;
  const int d = threadIdx.x;
  float A[16], h[16];
#pragma unroll
  for (int s = 0; s < 16; ++s) { A[s] = -__expf(A_log[d*16 + s]); h[s] = 0.f; }
  const float Dd = Dv[d];
  for (int l = 0; l < L; ++l) {
    const size_t row = (size_t)b * L + l;
    if (threadIdx.x < 32) {
      int s = threadIdx.x & 15;
      float v = dbl[row * lddbl + dtrank + ((threadIdx.x >> 4) ? 16 : 0) + s];
      if (threadIdx.x < 16) sB[s] = v; else sC[s] = v;
    }
    __syncthreads();
    float dtv = dt[row * D + d];
    float xv  = xc[row * D + d];
    float dx  = dtv * xv;
    float y = 0.f;
#pragma unroll
    for (int s = 0; s < 16; ++s) {
      h[s] = h[s] * __expf(dtv * A[s]) + dx * sB[s];
      y += h[s] * sC[s];
    }
    float zv = z[row * ldz + d];
    yout[row * D + d] = (y + xv * Dd) * (zv * sigmoidf_(zv));
    __syncthreads();
  }
}

// ---------- out = clip( xn + gelu( pr*BN_INV*g[c] + b[c] ), -10, 10 ) --
__global__ void bn_gelu_res(const float* __restrict__ xn, const float* __restrict__ pr,
                            const float* __restrict__ g, const float* __restrict__ bb,
                            float* __restrict__ out, int C, int total)
{
  int idx = blockIdx.x * blockDim.x + threadIdx.x;
  if (idx >= total) return;
  int c = idx % C;
  float v = pr[idx] * BN_INV * g[c] + bb[c];
  v = 0.5f * v * (1.0f + erff(v * 0.70710678118654752f));   // exact gelu
  v = xn[idx] + v;
  out[idx] = fminf(fmaxf(v, -10.f), 10.f);
}

// ---------- 3x3 conv (Cin=64, Cout=3, pad 1, NHWC in) + BN + sigmoid ---
__global__ __launch_bounds__(256) void conv3_bn_sig(
    const float* __restrict__ in, const float* __restrict__ wt,
    const float* __restrict__ cb, const float* __restrict__ g,
    const float* __restrict__ bb, float* __restrict__ out, int H, int W, int total)
{
  __shared__ float swt[3*64*9];
  for (int i = threadIdx.x; i < 3*64*9; i += 256) swt[i] = wt[i];
  __syncthreads();
  int idx = blockIdx.x * 256 + threadIdx.x;
  if (idx >= total) return;
  int w = idx % W;
  int h = (idx / W) % H;
  int b = idx / (H * W);
  float a0 = 0.f, a1 = 0.f, a2 = 0.f;
  for (int dy = 0; dy < 3; ++dy) {
    int hy = h + dy - 1;
    if (hy < 0 || hy >= H) continue;
    for (int dx = 0; dx < 3; ++dx) {
      int wx = w + dx - 1;
      if (wx < 0 || wx >= W) continue;
      const float* pix = in + ((size_t)b * H * W + hy * W + wx) * 64;
      int wo = dy * 3 + dx;
      for (int c = 0; c < 64; ++c) {
        float v = pix[c];
        a0 += v * swt[(0*64 + c) * 9 + wo];
        a1 += v * swt[(1*64 + c) * 9 + wo];
        a2 += v * swt[(2*64 + c) * 9 + wo];
      }
    }
  }
  float r[3] = {a0, a1, a2};
  size_t ob = ((size_t)b * H * W + h * W + w) * 3;
#pragma unroll
  for (int o = 0; o < 3; ++o) {
    float v = (r[o] + cb[o]) * BN_INV * g[o] + bb[o];
    out[ob + o] = sigmoidf_(v);
  }
}

// ---------- bilinear 2x resize (B,128,128,3) NHWC -> (B,3,256,256) NCHW
__global__ void resize_out(const float* __restrict__ in, float* __restrict__ out, int total)
{
  int idx = blockIdx.x * blockDim.x + threadIdx.x;
  if (idx >= total) return;
  int wo = idx % 256;
  int ho = (idx / 256) % 256;
  int o  = (idx / (256*256)) % 3;
  int b  = idx / (3*256*256);
  float sy = 0.5f * ho - 0.25f;
  float sx = 0.5f * wo - 0.25f;
  int y0 = (int)floorf(sy); float fy = sy - (float)y0;
  int x0 = (int)floorf(sx); float fx = sx - (float)x0;
  int y0c = min(max(y0, 0), 127), y1c = min(max(y0+1, 0), 127);
  int x0c = min(max(x0, 0), 127), x1c = min(max(x0+1, 0), 127);
  const float* base = in + (size_t)b * 128*128*3 + o;
  float v00 = base[(y0c*128 + x0c)*3], v01 = base[(y0c*128 + x1c)*3];
  float v10 = base[(y1c*128 + x0c)*3], v11 = base[(y1c*128 + x1c)*3];
  float top = v00 + (v01 - v00) * fx;
  float bot = v10 + (v11 - v10) * fx;
  out[idx] = top + (bot - top) * fy;
}

// ======================================================================
// Input index map (setup_inputs() dict order, nested dicts in insertion order)
// ======================================================================
enum {
  IN_C0=0, IN_C1, IN_C2, IN_UP1W, IN_UP1B,
  M1_NORM_G, M1_NORM_B, M1_IN_PROJ, M1_CONV_W, M1_CONV_B, M1_X_PROJ,
  M1_DT_W, M1_DT_B, M1_A_LOG, M1_D, M1_OUT_PROJ, M1_PROJ_W, M1_PROJ_B,
  M1_BN_G, M1_BN_B,
  IN_UP2W, IN_UP2B,
  M2_NORM_G, M2_NORM_B, M2_IN_PROJ, M2_CONV_W, M2_CONV_B, M2_X_PROJ,
  M2_DT_W, M2_DT_B, M2_A_LOG, M2_D, M2_OUT_PROJ, M2_PROJ_W, M2_PROJ_B,
  M2_BN_G, M2_BN_B,
  IN_OUTW, IN_OUTB, IN_OBN_G, IN_OBN_B
};

extern "C" void kernel_launch(void* const* d_in, const int* in_sizes, int n_in,
                              void* d_out, int out_size, void* d_ws, size_t ws_size,
                              hipStream_t stream) {
  (void)in_sizes; (void)n_in; (void)out_size; (void)ws_size;
  auto F = [&](int i){ return (const float*)d_in[i]; };

  // ---- workspace bump allocator (floats), sized for the level-2 shapes
  float* ws = (float*)d_ws;
  size_t off = 0;
  auto alloc = [&](size_t n){ float* p = ws + off; off += n; return p; };
  float* CSEQ  = alloc(1048576);   // c2 in seq layout
  float* WP    = alloc(131072);    // packed up-conv weights
  float* TMPU  = alloc(4194304);   // up-conv GEMM output
  float* AIN   = alloc(4194304);   // x = up + skip (seq)
  float* XN    = alloc(4194304);   // normalized+clipped
  float* XZ    = alloc(16777216);  // in_proj output (xi | z)
  float* XC    = alloc(8388608);   // conv1d+silu output
  float* DBLB  = alloc(2359296);   // x_proj output (dt_in | B | C)
  float* DTB   = alloc(8388608);   // softplus(dt)
  float* YOUT  = alloc(8388608);   // scan output (fused)
  float* YO    = alloc(4194304);   // out_proj output
  float* PR    = alloc(4194304);   // channel proj output
  float* BLK   = alloc(4194304);   // mamba block output
  float* STATS = alloc(16);        // per-batch mean/rstd
  float* CONV3 = alloc(196608);    // conv3x3 output NHWC

  auto gemm = [&](const float* A, const float* W, const float* bias, float* out,
                  int M, int N, int K, int lda, int ldo, int act){
    dim3 grid((N + 15) / 16, M / 64);
    gemm_xwt<<<grid, dim3(128), 0, stream>>>(A, W, bias, out, M, N, K, lda, ldo, act);
  };
  auto blks = [](int n){ return (n + 255) / 256; };
  const int B = 4;

  // ================= Level 1: up1 + mamba_block(m1, dim=128) ==========
  { int tot = B*256*1024;
    nchw_to_seq<<<blks(tot), 256, 0, stream>>>(F(IN_C2), CSEQ, 256, 1024, tot); }
  { int tot = 4*128*256;
    pack_upw<<<blks(tot), 256, 0, stream>>>(F(IN_UP1W), WP, 256, 128, tot); }
  gemm(CSEQ, WP, nullptr, TMPU, 4096, 512, 256, 256, 512, 0);
  { int tot = B*64*64*128;
    up_scatter<<<blks(tot), 256, 0, stream>>>(TMPU, F(IN_C1), F(IN_UP1B), AIN, 32, 32, 128, tot); }
  norm_stats<<<B, 256, 0, stream>>>(AIN, STATS, 524288);
  { int tot = B*524288;
    gn_clip<<<blks(tot), 256, 0, stream>>>(AIN, STATS, F(M1_NORM_G), F(M1_NORM_B), XN, 128, 524288, tot); }
  gemm(XN, F(M1_IN_PROJ), nullptr, XZ, 16384, 512, 128, 128, 512, 0);
  { int tot = B*4096*256;
    conv1d_silu<<<blks(tot), 256, 0, stream>>>(XZ, F(M1_CONV_W), F(M1_CONV_B), XC, 4096, 256, 512, tot); }
  gemm(XC, F(M1_X_PROJ), nullptr, DBLB, 16384, 40, 256, 256, 40, 0);
  gemm(DBLB, F(M1_DT_W), F(M1_DT_B), DTB, 16384, 256, 8, 40, 256, 1);
  mamba_scan<<<B, 256, 0, stream>>>(DTB, DBLB, XC, XZ + 256, F(M1_A_LOG), F(M1_D), YOUT,
                                    4096, 256, 8, 40, 512);
  gemm(YOUT, F(M1_OUT_PROJ), nullptr, YO, 16384, 128, 256, 256, 128, 0);
  gemm(YO, F(M1_PROJ_W), F(M1_PROJ_B), PR, 16384, 128, 128, 128, 128, 0);
  { int tot = 16384*128;
    bn_gelu_res<<<blks(tot), 256, 0, stream>>>(XN, PR, F(M1_BN_G), F(M1_BN_B), BLK, 128, tot); }

  // ================= Level 2: up2 + mamba_block(m2, dim=64) ===========
  { int tot = 4*64*128;
    pack_upw<<<blks(tot), 256, 0, stream>>>(F(IN_UP2W), WP, 128, 64, tot); }
  gemm(BLK, WP, nullptr, TMPU, 16384, 256, 128, 128, 256, 0);
  { int tot = B*128*128*64;
    up_scatter<<<blks(tot), 256, 0, stream>>>(TMPU, F(IN_C0), F(IN_UP2B), AIN, 64, 64, 64, tot); }
  norm_stats<<<B, 256, 0, stream>>>(AIN, STATS, 1048576);
  { int tot = B*1048576;
    gn_clip<<<blks(tot), 256, 0, stream>>>(AIN, STATS, F(M2_NORM_G), F(M2_NORM_B), XN, 64, 1048576, tot); }
  gemm(XN, F(M2_IN_PROJ), nullptr, XZ, 65536, 256, 64, 64, 256, 0);
  { int tot = B*16384*128;
    conv1d_silu<<<blks(tot), 256, 0, stream>>>(XZ, F(M2_CONV_W), F(M2_CONV_B), XC, 16384, 128, 256, tot); }
  gemm(XC, F(M2_X_PROJ), nullptr, DBLB, 65536, 36, 128, 128, 36, 0);
  gemm(DBLB, F(M2_DT_W), F(M2_DT_B), DTB, 65536, 128, 4, 36, 128, 1);
  mamba_scan<<<B, 128, 0, stream>>>(DTB, DBLB, XC, XZ + 128, F(M2_A_LOG), F(M2_D), YOUT,
                                    16384, 128, 4, 36, 256);
  gemm(YOUT, F(M2_OUT_PROJ), nullptr, YO, 65536, 64, 128, 128, 64, 0);
  gemm(YO, F(M2_PROJ_W), F(M2_PROJ_B), PR, 65536, 64, 64, 64, 64, 0);
  { int tot = 65536*64;
    bn_gelu_res<<<blks(tot), 256, 0, stream>>>(XN, PR, F(M2_BN_G), F(M2_BN_B), BLK, 64, tot); }

  // ================= Output head ======================================
  { int tot = B*128*128;
    conv3_bn_sig<<<blks(tot), 256, 0, stream>>>(BLK, F(IN_OUTW), F(IN_OUTB),
                                                F(IN_OBN_G), F(IN_OBN_B), CONV3, 128, 128, tot); }
  { int tot = B*3*256*256;
    resize_out<<<blks(tot), 256, 0, stream>>>(CONV3, (float*)d_out, tot); }
}